// GAT_61263413510492
// MI455X (gfx1250) — compile-verified
//
#include <hip/hip_runtime.h>
#include <math.h>

#define HEADS 4
#define HID 64
#define HC 256           // HEADS * HID
#define NEG_SLOPE 0.2f
#define EPS_F 1e-16f

typedef __attribute__((ext_vector_type(2))) float v2f;
typedef __attribute__((ext_vector_type(8))) float v8f;

// ---------- order-preserving float <-> uint mapping for atomic segment-max ----------
__device__ __forceinline__ unsigned f2ord(float f) {
    unsigned u = __float_as_uint(f);
    return (u & 0x80000000u) ? ~u : (u | 0x80000000u);
}
__device__ __forceinline__ float ord2f(unsigned u) {
    return (u & 0x80000000u) ? __uint_as_float(u & 0x7fffffffu) : __uint_as_float(~u);
}

// ---------- fp32 WMMA GEMM: C[M x NC] = A[M x K] @ B[K x NC] ----------
// One wave computes a 16x64 strip (four 16x16 f32 accumulators), K stepped by 4
// with V_WMMA_F32_16X16X4_F32. mode: 0 = raw, 1 = +bias, 2 = +bias then sigmoid.
__device__ __forceinline__ void epi_store(float* __restrict__ C, const float* __restrict__ bias,
                                          size_t rowoff, int col, int NC, int mode, float v) {
    if (col < NC) {
        if (mode >= 1) v += bias[col];
        if (mode == 2) v = 1.0f / (1.0f + __expf(-v));
        C[rowoff + col] = v;
    }
}

__global__ __launch_bounds__(256) void wmma_gemm_f32_kernel(
    const float* __restrict__ A, const float* __restrict__ B,
    float* __restrict__ C, const float* __restrict__ bias,
    int M, int K, int NC, int mode)
{
    const int wave = threadIdx.x >> 5;
    const int lane = threadIdx.x & 31;
    const int m0 = (blockIdx.x * 8 + wave) * 16;
    if (m0 >= M) return;                       // wave-uniform: EXEC stays all-1s
    const int n0 = blockIdx.y * 64;

    // A fragment (16x4 f32): lane<16 -> K=kb+0,kb+1 ; lane>=16 -> K=kb+2,kb+3
    const int aRow = m0 + (lane & 15);
    const int aK   = (lane >> 4) << 1;
    // B fragment (4x16 f32): mirrored K split, N = lane&15
    const int bN   = lane & 15;
    const int bK   = (lane >> 4) << 1;
    const int c    = n0 + bN;

    const float* Ap = A + (size_t)aRow * K + aK;     // advances by 4 floats per K-step
    const size_t stepB = (size_t)4 * NC;             // 4 B-rows per K-step

    v8f acc0 = {}, acc1 = {}, acc2 = {}, acc3 = {};

    if (n0 + 64 <= NC) {
        // -------- fast path: whole 64-wide strip in range, no predication --------
        const float* Bp = B + c + (size_t)bK * NC;   // row bK  , col c
        const float* Bq = Bp + NC;                   // row bK+1, col c
        #pragma unroll 2
        for (int kb = 0; kb < K; kb += 4) {
            v2f a = *(const v2f*)Ap;                 // global_load_b64
            v2f b0 = { Bp[0],  Bq[0]  };
            v2f b1 = { Bp[16], Bq[16] };
            v2f b2 = { Bp[32], Bq[32] };
            v2f b3 = { Bp[48], Bq[48] };
            Ap += 4; Bp += stepB; Bq += stepB;
            acc0 = __builtin_amdgcn_wmma_f32_16x16x4_f32(false, a, false, b0, (short)0, acc0, false, false);
            acc1 = __builtin_amdgcn_wmma_f32_16x16x4_f32(false, a, false, b1, (short)0, acc1, false, false);
            acc2 = __builtin_amdgcn_wmma_f32_16x16x4_f32(false, a, false, b2, (short)0, acc2, false, false);
            acc3 = __builtin_amdgcn_wmma_f32_16x16x4_f32(false, a, false, b3, (short)0, acc3, false, false);
        }
    } else {
        // -------- edge path (NC not multiple of 64): clamped cols + 0/1 masks,
        // loads stay unpredicated so no exec-mask juggling in the loop --------
        const int   c0 = min(c,      NC - 1), c1 = min(c + 16, NC - 1);
        const int   c2 = min(c + 32, NC - 1), c3 = min(c + 48, NC - 1);
        const float w0 = (c      < NC) ? 1.0f : 0.0f;
        const float w1 = (c + 16 < NC) ? 1.0f : 0.0f;
        const float w2 = (c + 32 < NC) ? 1.0f : 0.0f;
        const float w3 = (c + 48 < NC) ? 1.0f : 0.0f;
        const float* Bp = B + (size_t)bK * NC;
        const float* Bq = Bp + NC;
        #pragma unroll 2
        for (int kb = 0; kb < K; kb += 4) {
            v2f a = *(const v2f*)Ap;
            v2f b0 = { Bp[c0] * w0, Bq[c0] * w0 };
            v2f b1 = { Bp[c1] * w1, Bq[c1] * w1 };
            v2f b2 = { Bp[c2] * w2, Bq[c2] * w2 };
            v2f b3 = { Bp[c3] * w3, Bq[c3] * w3 };
            Ap += 4; Bp += stepB; Bq += stepB;
            acc0 = __builtin_amdgcn_wmma_f32_16x16x4_f32(false, a, false, b0, (short)0, acc0, false, false);
            acc1 = __builtin_amdgcn_wmma_f32_16x16x4_f32(false, a, false, b1, (short)0, acc1, false, false);
            acc2 = __builtin_amdgcn_wmma_f32_16x16x4_f32(false, a, false, b2, (short)0, acc2, false, false);
            acc3 = __builtin_amdgcn_wmma_f32_16x16x4_f32(false, a, false, b3, (short)0, acc3, false, false);
        }
    }

    // C/D layout: VGPR r holds M = r (lanes 0-15) / M = 8+r (lanes 16-31), N = lane&15
    const int mbase = m0 + ((lane >> 4) << 3);
    const int colb  = n0 + (lane & 15);
    #pragma unroll
    for (int r = 0; r < 8; ++r) {
        size_t rowoff = (size_t)(mbase + r) * NC;
        epi_store(C, bias, rowoff, colb,      NC, mode, acc0[r]);
        epi_store(C, bias, rowoff, colb + 16, NC, mode, acc1[r]);
        epi_store(C, bias, rowoff, colb + 32, NC, mode, acc2[r]);
        epi_store(C, bias, rowoff, colb + 48, NC, mode, acc3[r]);
    }
}

// ---------- per-(node,head) attention logits a_s/a_d; also init max/denom ----------
__global__ __launch_bounds__(256) void scores_kernel(
    const float* __restrict__ H, const float* __restrict__ att_src,
    const float* __restrict__ att_dst, float* __restrict__ a_s, float* __restrict__ a_d,
    unsigned* __restrict__ mmax, float* __restrict__ denom, int NH)
{
    const int gw   = (int)((blockIdx.x * 256 + threadIdx.x) >> 5);  // one wave per (node,head)
    const int lane = threadIdx.x & 31;
    if (gw >= NH) return;
    const int n = gw >> 2, h = gw & 3;
    const float* hp = H + (size_t)n * HC + h * HID;
    const float x0 = hp[lane], x1 = hp[lane + 32];
    float s = x0 * att_src[h * HID + lane] + x1 * att_src[h * HID + lane + 32];
    float d = x0 * att_dst[h * HID + lane] + x1 * att_dst[h * HID + lane + 32];
    #pragma unroll
    for (int off = 16; off > 0; off >>= 1) {
        s += __shfl_xor(s, off, 32);
        d += __shfl_xor(d, off, 32);
    }
    if (lane == 0) {
        a_s[gw] = s; a_d[gw] = d;
        mmax[gw] = f2ord(-INFINITY);
        denom[gw] = 0.0f;
    }
}

__device__ __forceinline__ void edge_src_dst(const int* __restrict__ ei, int E, int eid,
                                             int& src, int& dst) {
    if (eid < E) { src = ei[eid]; dst = ei[E + eid]; }
    else         { src = dst = eid - E; }       // appended self-loops
}

// ---------- pass 1: segment max over dst ----------
__global__ __launch_bounds__(256) void edge_max_kernel(
    const int* __restrict__ ei, int E, int N,
    const float* __restrict__ a_s, const float* __restrict__ a_d, unsigned* __restrict__ mmax)
{
    const int t = blockIdx.x * 256 + threadIdx.x;
    if (t >= (E + N) * 4) return;
    const int eid = t >> 2, h = t & 3;
    int src, dst; edge_src_dst(ei, E, eid, src, dst);
    float e = a_s[src * 4 + h] + a_d[dst * 4 + h];
    e = (e > 0.0f) ? e : NEG_SLOPE * e;         // leaky_relu
    atomicMax(mmax + dst * 4 + h, f2ord(e));    // global_atomic_max_u32, L2-resident
}

// ---------- pass 2: exp(e - max) and segment sum ----------
__global__ __launch_bounds__(256) void edge_sum_kernel(
    const int* __restrict__ ei, int E, int N,
    const float* __restrict__ a_s, const float* __restrict__ a_d,
    const unsigned* __restrict__ mmax, float* __restrict__ denom, float* __restrict__ ew)
{
    const int t = blockIdx.x * 256 + threadIdx.x;
    if (t >= (E + N) * 4) return;
    const int eid = t >> 2, h = t & 3;
    int src, dst; edge_src_dst(ei, E, eid, src, dst);
    float e = a_s[src * 4 + h] + a_d[dst * 4 + h];
    e = (e > 0.0f) ? e : NEG_SLOPE * e;
    const float ex = __expf(e - ord2f(mmax[dst * 4 + h]));
    ew[t] = ex;
    atomicAdd(denom + dst * 4 + h, ex);         // global_atomic_add_f32
}

// ---------- pass 3: out[dst] += alpha * h[src]  (one wave per edge, 8 ch/lane) ----------
__global__ __launch_bounds__(256) void edge_agg_kernel(
    const int* __restrict__ ei, int E, int N, const float* __restrict__ H,
    const float* __restrict__ ew, const float* __restrict__ denom, float* __restrict__ out)
{
    const int gw   = (int)((blockIdx.x * 256 + threadIdx.x) >> 5);
    const int lane = threadIdx.x & 31;
    if (gw >= E + N) return;
    int src, dst; edge_src_dst(ei, E, gw, src, dst);
    const int head = lane >> 3;                 // lane covers channels [lane*8, lane*8+8)
    const float alpha = ew[(size_t)gw * 4 + head] / (denom[dst * 4 + head] + EPS_F);
    const float4* hs = (const float4*)(H + (size_t)src * HC) + (lane << 1);
    const float4 v0 = hs[0], v1 = hs[1];        // 2x global_load_b128, L2-resident
    float* op = out + (size_t)dst * HC + (lane << 3);
    atomicAdd(op + 0, alpha * v0.x);
    atomicAdd(op + 1, alpha * v0.y);
    atomicAdd(op + 2, alpha * v0.z);
    atomicAdd(op + 3, alpha * v0.w);
    atomicAdd(op + 4, alpha * v1.x);
    atomicAdd(op + 5, alpha * v1.y);
    atomicAdd(op + 6, alpha * v1.z);
    atomicAdd(op + 7, alpha * v1.w);
}

__global__ __launch_bounds__(256) void zero_kernel(float* __restrict__ p, int n) {
    const int i = blockIdx.x * 256 + threadIdx.x;
    if (i < n) p[i] = 0.0f;
}

__global__ __launch_bounds__(256) void bias_relu_kernel(float* __restrict__ p,
                                                        const float* __restrict__ b, int n) {
    const int i = blockIdx.x * 256 + threadIdx.x;
    if (i < n) {
        const float v = p[i] + b[i & (HC - 1)];
        p[i] = fmaxf(v, 0.0f);
    }
}

extern "C" void kernel_launch(void* const* d_in, const int* in_sizes, int n_in,
                              void* d_out, int out_size, void* d_ws, size_t ws_size,
                              hipStream_t stream)
{
    const float* x   = (const float*)d_in[0];
    const int*   ei  = (const int*)d_in[1];
    const float* W1  = (const float*)d_in[2];
    const float* as1 = (const float*)d_in[3];
    const float* ad1 = (const float*)d_in[4];
    const float* b1  = (const float*)d_in[5];
    const float* W2  = (const float*)d_in[6];
    const float* as2 = (const float*)d_in[7];
    const float* ad2 = (const float*)d_in[8];
    const float* b2  = (const float*)d_in[9];
    const float* Wm1 = (const float*)d_in[10];
    const float* bm1 = (const float*)d_in[11];
    const float* Wm2 = (const float*)d_in[12];
    const float* bm2 = (const float*)d_in[13];

    const int N    = in_sizes[0] / 128;
    const int E    = in_sizes[1] / 2;
    const int Etot = E + N;
    const int NH   = N * HEADS;

    // ---- workspace layout (floats), ~120 MB ----
    float* ws = (float*)d_ws;
    size_t o = 0;
    float*    P0    = ws + o; o += (size_t)N * HC;       // pre-aggregation features h
    float*    P1    = ws + o; o += (size_t)N * HC;       // aggregated output / next input
    float*    a_s   = ws + o; o += (size_t)NH;
    float*    a_d   = ws + o; o += (size_t)NH;
    unsigned* mmax  = (unsigned*)(ws + o); o += (size_t)NH;
    float*    denom = ws + o; o += (size_t)NH;
    float*    ew    = ws + o; o += (size_t)Etot * HEADS;

    const dim3 blk(256);
    auto gemm_grid = [](int M, int NC) {
        return dim3((unsigned)(((M + 15) / 16 + 7) / 8), (unsigned)((NC + 63) / 64));
    };
    const int gScores = (NH * 32 + 255) / 256;
    const int gEdgeH  = (Etot * 4 + 255) / 256;
    const int gEdgeW  = (Etot * 32 + 255) / 256;
    const int gFeat   = (N * HC + 255) / 256;

    // ================= conv1: GAT(x[N,128] -> P1[N,256]) =================
    wmma_gemm_f32_kernel<<<gemm_grid(N, HC), blk, 0, stream>>>(x, W1, P0, nullptr, N, 128, HC, 0);
    scores_kernel<<<gScores, blk, 0, stream>>>(P0, as1, ad1, a_s, a_d, mmax, denom, NH);
    edge_max_kernel<<<gEdgeH, blk, 0, stream>>>(ei, E, N, a_s, a_d, mmax);
    edge_sum_kernel<<<gEdgeH, blk, 0, stream>>>(ei, E, N, a_s, a_d, mmax, denom, ew);
    zero_kernel<<<gFeat, blk, 0, stream>>>(P1, N * HC);
    edge_agg_kernel<<<gEdgeW, blk, 0, stream>>>(ei, E, N, P0, ew, denom, P1);
    bias_relu_kernel<<<gFeat, blk, 0, stream>>>(P1, b1, N * HC);

    // ================= conv2: GAT(P1[N,256] -> P1[N,256]) =================
    wmma_gemm_f32_kernel<<<gemm_grid(N, HC), blk, 0, stream>>>(P1, W2, P0, nullptr, N, HC, HC, 0);
    scores_kernel<<<gScores, blk, 0, stream>>>(P0, as2, ad2, a_s, a_d, mmax, denom, NH);
    edge_max_kernel<<<gEdgeH, blk, 0, stream>>>(ei, E, N, a_s, a_d, mmax);
    edge_sum_kernel<<<gEdgeH, blk, 0, stream>>>(ei, E, N, a_s, a_d, mmax, denom, ew);
    zero_kernel<<<gFeat, blk, 0, stream>>>(P1, N * HC);   // P1 input already consumed by GEMM
    edge_agg_kernel<<<gEdgeW, blk, 0, stream>>>(ei, E, N, P0, ew, denom, P1);
    bias_relu_kernel<<<gFeat, blk, 0, stream>>>(P1, b2, N * HC);

    // ================= post_mp: Linear(256->64)+b, Linear(64->121)+b, sigmoid ===========
    wmma_gemm_f32_kernel<<<gemm_grid(N, HID), blk, 0, stream>>>(P1, Wm1, P0, bm1, N, HC, HID, 1);
    wmma_gemm_f32_kernel<<<gemm_grid(N, 121), blk, 0, stream>>>(P0, Wm2, (float*)d_out, bm2,
                                                                N, HID, 121, 2);
}